// GTLayer_4355096839072
// MI455X (gfx1250) — compile-verified
//
#include <hip/hip_runtime.h>
#include <hip/hip_bf16.h>

typedef float v2f __attribute__((ext_vector_type(2)));
typedef float v8f __attribute__((ext_vector_type(8)));

#define EMB 64
#define HEADS 4
#define HEAD_DIM 16

// ---------------------------------------------------------------------------
// Kernel 0: zero a float buffer
// ---------------------------------------------------------------------------
__global__ void zero_f32(float* __restrict__ p, int n) {
    int i = blockIdx.x * blockDim.x + threadIdx.x;
    if (i < n) p[i] = 0.0f;
}

// ---------------------------------------------------------------------------
// Kernel 1: Q/K/V = embeds @ {qW,kW,vW}  using V_WMMA_F32_16X16X4_F32
// One wave computes one 16(node)x16(col) tile for all three weight matrices,
// sharing the A fragment across the three WMMAs per K-step.
// Block = 128 threads = 4 waves = 4 column tiles (4*16 = 64 = EMB).
// ---------------------------------------------------------------------------
__global__ __launch_bounds__(128)
void qkv_wmma(const float* __restrict__ embeds,
              const float* __restrict__ qW,
              const float* __restrict__ kW,
              const float* __restrict__ vW,
              float* __restrict__ Q,
              float* __restrict__ K,
              float* __restrict__ V,
              int nNodes)
{
    const int wave = threadIdx.x >> 5;          // 0..3 -> column tile
    const int lane = threadIdx.x & 31;
    const int half = lane >> 4;                 // 0 or 1
    const int lid  = lane & 15;
    const int nodeBase = blockIdx.x * 16;
    const int colBase  = wave * 16;

    // A-fragment source row for this lane (clamped so EXEC stays all-1s)
    int aRow = nodeBase + lid;
    if (aRow >= nNodes) aRow = nNodes - 1;
    const float* __restrict__ aPtr = embeds + (size_t)aRow * EMB;

    v8f accQ = {0.f,0.f,0.f,0.f,0.f,0.f,0.f,0.f};
    v8f accK = accQ;
    v8f accV = accQ;

    #pragma unroll
    for (int k = 0; k < EMB; k += 4) {
        const int kk = k + 2 * half;            // this lane's K pair
        // A 16x4 fragment: lane holds E[aRow][kk], E[aRow][kk+1]  (8B aligned)
        v2f a = *reinterpret_cast<const v2f*>(aPtr + kk);

        // B 4x16 fragments: lane n holds W[kk][col], W[kk+1][col]
        const int col = colBase + lid;
        v2f bq, bk, bv;
        bq.x = qW[(size_t)kk * EMB + col];      bq.y = qW[(size_t)(kk + 1) * EMB + col];
        bk.x = kW[(size_t)kk * EMB + col];      bk.y = kW[(size_t)(kk + 1) * EMB + col];
        bv.x = vW[(size_t)kk * EMB + col];      bv.y = vW[(size_t)(kk + 1) * EMB + col];

        accQ = __builtin_amdgcn_wmma_f32_16x16x4_f32(false, a, false, bq, (short)0, accQ, false, false);
        accK = __builtin_amdgcn_wmma_f32_16x16x4_f32(false, a, false, bk, (short)0, accK, false, false);
        accV = __builtin_amdgcn_wmma_f32_16x16x4_f32(false, a, false, bv, (short)0, accV, false, false);
    }

    // D layout: VGPR r, lanes 0-15 -> M=r, lanes 16-31 -> M=r+8; N = lid
    #pragma unroll
    for (int r = 0; r < 8; ++r) {
        const int row = nodeBase + r + 8 * half;
        if (row < nNodes) {
            const size_t o = (size_t)row * EMB + colBase + lid;
            Q[o] = accQ[r];
            K[o] = accK[r];
            V[o] = accV[r];
        }
    }
}

// ---------------------------------------------------------------------------
// Kernel 2: per-(edge,head) attention score
//   att = clip(q . k, -10, 10); e = exp(att); expAtt[e,h] = e;
//   denom[row,h] += e   (f32 atomic, L2-resident 800KB table)
// ---------------------------------------------------------------------------
__global__ __launch_bounds__(256)
void edge_att(const float4* __restrict__ Q4,
              const float4* __restrict__ K4,
              const int* __restrict__ rows,
              const int* __restrict__ cols,
              float* __restrict__ expAtt,
              float* __restrict__ denom,
              int nEdges)
{
    int t = blockIdx.x * blockDim.x + threadIdx.x;
    if (t >= nEdges * HEADS) return;
    const int e = t >> 2;
    const int h = t & 3;
    const int r = rows[e];
    const int c = cols[e];

    const float4* __restrict__ q = Q4 + (size_t)r * (EMB / 4) + h * 4;
    const float4* __restrict__ k = K4 + (size_t)c * (EMB / 4) + h * 4;

    float s = 0.0f;
    #pragma unroll
    for (int i = 0; i < 4; ++i) {
        float4 a = q[i];
        float4 b = k[i];
        s += a.x * b.x + a.y * b.y + a.z * b.z + a.w * b.w;
    }
    s = fminf(10.0f, fmaxf(-10.0f, s));
    float ex = __expf(s);
    expAtt[t] = ex;
    atomicAdd(&denom[(size_t)r * HEADS + h], ex);
}

// ---------------------------------------------------------------------------
// Kernel 3: per-(edge,head) normalize + weighted scatter-add of V into out
// out is 12.8MB -> atomics land in L2.
// ---------------------------------------------------------------------------
__global__ __launch_bounds__(256)
void edge_agg(const float4* __restrict__ V4,
              const int* __restrict__ rows,
              const int* __restrict__ cols,
              const float* __restrict__ expAtt,
              const float* __restrict__ denom,
              float* __restrict__ out,
              int nEdges)
{
    int t = blockIdx.x * blockDim.x + threadIdx.x;
    if (t >= nEdges * HEADS) return;
    const int e = t >> 2;
    const int h = t & 3;
    const int r = rows[e];
    const int c = cols[e];

    const float att = expAtt[t] / (denom[(size_t)r * HEADS + h] + 1e-8f);

    #pragma unroll
    for (int i = 0; i < 4; ++i) {
        float4 v = V4[(size_t)c * (EMB / 4) + h * 4 + i];
        float* o = out + (size_t)r * EMB + h * HEAD_DIM + i * 4;
        atomicAdd(o + 0, att * v.x);
        atomicAdd(o + 1, att * v.y);
        atomicAdd(o + 2, att * v.z);
        atomicAdd(o + 3, att * v.w);
    }
}

// ---------------------------------------------------------------------------
extern "C" void kernel_launch(void* const* d_in, const int* in_sizes, int n_in,
                              void* d_out, int out_size, void* d_ws, size_t ws_size,
                              hipStream_t stream) {
    const float* embeds = (const float*)d_in[0];
    const int*   rows   = (const int*)d_in[1];
    const int*   cols   = (const int*)d_in[2];
    const float* qW     = (const float*)d_in[3];
    const float* kW     = (const float*)d_in[4];
    const float* vW     = (const float*)d_in[5];
    float* out = (float*)d_out;

    const int nNodes = in_sizes[0] / EMB;   // 50000
    const int nEdges = in_sizes[1];         // 800000

    // Workspace layout (floats): Q | K | V | denom | expAtt
    float* ws     = (float*)d_ws;
    float* Q      = ws;
    float* K      = Q + (size_t)nNodes * EMB;
    float* V      = K + (size_t)nNodes * EMB;
    float* denom  = V + (size_t)nNodes * EMB;
    float* expAtt = denom + (size_t)nNodes * HEADS;

    // 1) zero the accumulation targets
    {
        int n = nNodes * EMB;
        zero_f32<<<(n + 255) / 256, 256, 0, stream>>>(out, n);
        int m = nNodes * HEADS;
        zero_f32<<<(m + 255) / 256, 256, 0, stream>>>(denom, m);
    }

    // 2) node-level Q/K/V projection via f32 WMMA
    {
        int blocks = (nNodes + 15) / 16;
        qkv_wmma<<<blocks, 128, 0, stream>>>(embeds, qW, kW, vW, Q, K, V, nNodes);
    }

    // 3) edge attention scores + segment-sum denominator
    {
        int work = nEdges * HEADS;
        edge_att<<<(work + 255) / 256, 256, 0, stream>>>(
            (const float4*)Q, (const float4*)K, rows, cols, expAtt, denom, nEdges);
    }

    // 4) normalize + scatter-add aggregation
    {
        int work = nEdges * HEADS;
        edge_agg<<<(work + 255) / 256, 256, 0, stream>>>(
            (const float4*)V, rows, cols, expAtt, denom, out, nEdges);
    }
}